// ChebNet_85985245266266
// MI455X (gfx1250) — compile-verified
//
#include <hip/hip_runtime.h>
#include <math.h>

typedef float v2f __attribute__((ext_vector_type(2)));
typedef float v8f __attribute__((ext_vector_type(8)));

#define DIN  64
#define HID  32
#define DOUT 40

// ---------------------------------------------------------------- utilities
__global__ void k_zero(float* __restrict__ p, int n) {
    int i = blockIdx.x * blockDim.x + threadIdx.x;
    if (i < n) p[i] = 0.0f;
}

// deg[dst] += 1 per edge (fp32 exact up to 2^24 > E)
__global__ void k_degree(const long long* __restrict__ ei, float* __restrict__ deg, int E) {
    int e = blockIdx.x * blockDim.x + threadIdx.x;
    if (e < E) {
        int d = (int)ei[(long long)E + e];
        atomicAdd(&deg[d], 1.0f);
    }
}

__global__ void k_dis(const float* __restrict__ deg, float* __restrict__ dis, int n) {
    int i = blockIdx.x * blockDim.x + threadIdx.x;
    if (i < n) {
        float d = deg[i];
        dis[i] = (d > 0.0f) ? rsqrtf(fmaxf(d, 1.0f)) : 0.0f;
    }
}

// ------------------------------------------------- GEMM1: x @ [W0_1 | W1_1]
// x: N x 64, W0_1/W1_1: 64 x 32. Output: xW0 (N x 32) and t = x@W1_1 (N x 32).
// Block = 128 threads = 4 waves; each wave owns one 16-wide column tile of the
// concatenated 64-wide output. One 16-row M-tile per block.
__global__ __launch_bounds__(128)
void k_gemm1(const float* __restrict__ x,
             const float* __restrict__ W0, const float* __restrict__ W1,
             float* __restrict__ xW0, float* __restrict__ t, int nrows) {
    const int wave = threadIdx.x >> 5;       // 0..3 -> column tile
    const int lane = threadIdx.x & 31;
    const int m0   = blockIdx.x * 16;
    const int rowA = min(m0 + (lane & 15), nrows - 1);   // clamp keeps EXEC full
    const int koff = (lane >> 4) * 2;        // 0 or 2
    const int nloc = lane & 15;
    const int gcol = wave * 16 + nloc;       // 0..63 in concatenated output
    const float* __restrict__ W = (gcol < HID) ? W0 : W1;
    const int wcol = (gcol < HID) ? gcol : (gcol - HID);

    v8f acc = {};
    #pragma unroll
    for (int k0 = 0; k0 < DIN; k0 += 4) {
        v2f a, b;
        a.x = x[rowA * DIN + k0 + koff];
        a.y = x[rowA * DIN + k0 + koff + 1];
        b.x = W[(k0 + koff)     * HID + wcol];
        b.y = W[(k0 + koff + 1) * HID + wcol];
        acc = __builtin_amdgcn_wmma_f32_16x16x4_f32(
            false, a, false, b, (short)0, acc, false, false);
    }

    float* __restrict__ out = (wave < 2) ? xW0 : t;
    const int ocol  = (wave & 1) * 16 + nloc;            // 0..31 in 32-wide out
    const int rbase = m0 + (lane >> 4) * 8;
    #pragma unroll
    for (int r = 0; r < 8; ++r) {
        int row = rbase + r;
        if (row < nrows) out[row * HID + ocol] = acc[r];
    }
}

// ------------------------------------------------------- edge scatter (prop)
// one wave per edge, lane = channel (32-wide features)
__global__ __launch_bounds__(256)
void k_scatter(const long long* __restrict__ ei, const float* __restrict__ dis,
               const float* __restrict__ feat, float* __restrict__ acc, int E) {
    const int lane = threadIdx.x & 31;
    const int e = blockIdx.x * 8 + (threadIdx.x >> 5);
    if (e >= E) return;
    const int s = (int)ei[e];
    const int d = (int)ei[(long long)E + e];
    const float w = -dis[s] * dis[d];
    atomicAdd(&acc[d * HID + lane], w * feat[s * HID + lane]);
}

// h = relu(xW0 + p1 + b1), written in place over p1 (bufC); also zero q (bufA)
__global__ void k_relu_bias(const float* __restrict__ xW0, const float* __restrict__ b1,
                            float* __restrict__ hc, float* __restrict__ qzero, int n) {
    int i = blockIdx.x * blockDim.x + threadIdx.x;
    if (i < n) {
        float v = xW0[i] + hc[i] + b1[i & (HID - 1)];
        hc[i] = v > 0.0f ? v : 0.0f;
        qzero[i] = 0.0f;
    }
}

// ----------------------------------- GEMM2: [h | q] @ [[W0_2],[W1_2]] + b2
// A: N x 64 (h then q), B: 64 x 40, out: N x 40 (pre-softmax) into d_out.
// Block = 96 threads = 3 waves (column tiles cover 48 >= 40 cols).
__global__ __launch_bounds__(96)
void k_gemm2(const float* __restrict__ h, const float* __restrict__ q,
             const float* __restrict__ W0, const float* __restrict__ W1,
             const float* __restrict__ b2, float* __restrict__ out, int nrows) {
    const int wave = threadIdx.x >> 5;       // 0..2
    const int lane = threadIdx.x & 31;
    const int m0   = blockIdx.x * 16;
    const int rowA = min(m0 + (lane & 15), nrows - 1);
    const int koff = (lane >> 4) * 2;
    const int nloc = lane & 15;
    const int gcol = wave * 16 + nloc;       // 0..47, valid < 40

    v8f acc = {};
    #pragma unroll
    for (int k0 = 0; k0 < 2 * HID; k0 += 4) {
        const int ka = k0 + koff;            // even -> ka, ka+1 never straddle 32
        v2f a, b;
        a.x = (ka     < HID) ? h[rowA * HID + ka]          : q[rowA * HID + ka - HID];
        a.y = (ka + 1 < HID) ? h[rowA * HID + ka + 1]      : q[rowA * HID + ka + 1 - HID];
        float bx = 0.0f, by = 0.0f;
        if (gcol < DOUT) {
            bx = (ka     < HID) ? W0[ka * DOUT + gcol]       : W1[(ka - HID) * DOUT + gcol];
            by = (ka + 1 < HID) ? W0[(ka + 1) * DOUT + gcol] : W1[(ka + 1 - HID) * DOUT + gcol];
        }
        b.x = bx; b.y = by;
        acc = __builtin_amdgcn_wmma_f32_16x16x4_f32(
            false, a, false, b, (short)0, acc, false, false);
    }

    if (gcol < DOUT) {
        const float bias = b2[gcol];
        const int rbase = m0 + (lane >> 4) * 8;
        #pragma unroll
        for (int r = 0; r < 8; ++r) {
            int row = rbase + r;
            if (row < nrows) out[row * DOUT + gcol] = acc[r] + bias;
        }
    }
}

// ------------------------------------------------ in-place row log_softmax
// one wave per row; lane covers col and col+32
__global__ __launch_bounds__(256)
void k_logsoftmax(float* __restrict__ o, int nrows) {
    const int lane = threadIdx.x & 31;
    const int row = blockIdx.x * 8 + (threadIdx.x >> 5);
    if (row >= nrows) return;
    float* __restrict__ rp = o + (long long)row * DOUT;
    const float v0 = rp[lane];
    const float v1 = (lane < DOUT - 32) ? rp[32 + lane] : -INFINITY;
    float m = fmaxf(v0, v1);
    #pragma unroll
    for (int off = 16; off > 0; off >>= 1)
        m = fmaxf(m, __shfl_xor(m, off, 32));
    float s = __expf(v0 - m) + ((lane < DOUT - 32) ? __expf(v1 - m) : 0.0f);
    #pragma unroll
    for (int off = 16; off > 0; off >>= 1)
        s += __shfl_xor(s, off, 32);
    const float lse = m + __logf(s);
    rp[lane] = v0 - lse;
    if (lane < DOUT - 32) rp[32 + lane] = v1 - lse;
}

// ---------------------------------------------------------------- launcher
extern "C" void kernel_launch(void* const* d_in, const int* in_sizes, int n_in,
                              void* d_out, int out_size, void* d_ws, size_t ws_size,
                              hipStream_t stream) {
    const float*      x    = (const float*)d_in[0];
    const long long*  ei   = (const long long*)d_in[1];   // int64 edge_index (2,E)
    const float*      W0_1 = (const float*)d_in[2];
    const float*      W1_1 = (const float*)d_in[3];
    const float*      b1   = (const float*)d_in[4];
    const float*      W0_2 = (const float*)d_in[5];
    const float*      W1_2 = (const float*)d_in[6];
    const float*      b2   = (const float*)d_in[7];
    float*            out  = (float*)d_out;

    const int N = in_sizes[0] / DIN;
    const int E = in_sizes[1] / 2;

    float* ws   = (float*)d_ws;
    float* deg  = ws;                          // N
    float* dis  = ws + (size_t)N;              // N
    float* bufA = ws + 2 * (size_t)N;          // N*32 : t = x@W1_1, later q
    float* bufB = bufA + 32 * (size_t)N;       // N*32 : xW0
    float* bufC = bufB + 32 * (size_t)N;       // N*32 : p1, later h

    const int T = 256;
    const int nMt = (N + 15) / 16;

    k_zero<<<(N + T - 1) / T, T, 0, stream>>>(deg, N);
    k_zero<<<(32 * N + T - 1) / T, T, 0, stream>>>(bufC, 32 * N);
    k_degree<<<(E + T - 1) / T, T, 0, stream>>>(ei, deg, E);
    k_dis<<<(N + T - 1) / T, T, 0, stream>>>(deg, dis, N);

    // layer 1: project first (prop commutes with matmul) -> 32-wide scatter
    k_gemm1<<<nMt, 128, 0, stream>>>(x, W0_1, W1_1, bufB, bufA, N);
    k_scatter<<<(E + 7) / 8, 256, 0, stream>>>(ei, dis, bufA, bufC, E);
    k_relu_bias<<<(32 * N + T - 1) / T, T, 0, stream>>>(bufB, b1, bufC, bufA, 32 * N);

    // layer 2: scatter 32-wide h, then project [h|q] -> 40-wide output
    k_scatter<<<(E + 7) / 8, 256, 0, stream>>>(ei, dis, bufC, bufA, E);
    k_gemm2<<<nMt, 96, 0, stream>>>(bufC, bufA, W0_2, W1_2, b2, out, N);
    k_logsoftmax<<<(N + 7) / 8, 256, 0, stream>>>(out, N);
}